// TagGraphGNN_89258010345478
// MI455X (gfx1250) — compile-verified
//
#include <hip/hip_runtime.h>
#include <hip/hip_bf16.h>

// ---------------------------------------------------------------------------
// TagGraphGNN on MI455X (gfx1250): 2-layer GraphSAGE + BN/ReLU + L2 norm.
// Edge scatter dominates (~1.7 GB HBM traffic). Dense GEMMs run on the CDNA5
// fp32 matrix pipe (v_wmma_f32_16x16x4_f32) with 80-row register blocking so
// weight reloads (L2) drop to ~640 MB total, well under the scatter cost.
// ---------------------------------------------------------------------------

#define N_NODES 50000
#define N_EDGES 800000
#define D       256
#define BN_EPS  1e-5f
#define NRM_EPS 1e-12f

#define GEMM_ROWS 80            // rows per block (5 strips of 16); 50000/80=625

typedef float v2f __attribute__((ext_vector_type(2)));
typedef float v8f __attribute__((ext_vector_type(8)));

// ---------------- degree counts: cnt[dst] += 1 ------------------------------
__global__ __launch_bounds__(256) void count_kernel(const int* __restrict__ ei,
                                                    float* __restrict__ cnt) {
    int e = blockIdx.x * 256 + threadIdx.x;
    if (e < N_EDGES) atomicAdd(&cnt[ei[N_EDGES + e]], 1.0f);
}

// ---------------- scatter: s[dst,:] += feat[src,:] --------------------------
// 4 edges per 256-thread block; each 64-lane group does one edge as float4s.
__global__ __launch_bounds__(256) void scatter_kernel(const float* __restrict__ feat,
                                                      const int* __restrict__ ei,
                                                      float* __restrict__ s) {
    int e = blockIdx.x * 4 + (threadIdx.x >> 6);
    int t = threadIdx.x & 63;
    int src = ei[e];
    int dst = ei[N_EDGES + e];
    const float4 v = ((const float4*)(feat + (size_t)src * D))[t];
    float* o = s + (size_t)dst * D + (t << 2);
    atomicAdd(o + 0, v.x);
    atomicAdd(o + 1, v.y);
    atomicAdd(o + 2, v.z);
    atomicAdd(o + 3, v.w);
}

// ---------------- mean aggregate: s /= cnt (0 if cnt==0) --------------------
__global__ __launch_bounds__(256) void agg_kernel(float* __restrict__ s,
                                                  const float* __restrict__ cnt) {
    int idx = blockIdx.x * 256 + threadIdx.x;       // float4 group id
    int row = idx >> 6;                             // 64 groups per row
    float c = cnt[row];
    float inv = c > 0.0f ? 1.0f / c : 0.0f;
    float4 v = ((float4*)s)[idx];
    v.x *= inv; v.y *= inv; v.z *= inv; v.w *= inv;
    ((float4*)s)[idx] = v;
}

// ---------------- fused SAGE GEMM: out = A1*W1^T + A2*W2^T + b --------------
// A1,A2: [M,256] row-major. W: [256 out,256 in] row-major (B[k][n] = W[n][k]).
// Block = 8 waves, 80-row strip (5 M-tiles of 16); wave w owns 32 columns
// [32w, 32w+32) as two 16x16 tiles. K-step = 4 (fp32 WMMA). Each B fragment
// load is amortized over 5 M-tiles (register blocking).
__global__ __launch_bounds__(256) void sage_gemm_kernel(const float* __restrict__ A1,
                                                        const float* __restrict__ W1,
                                                        const float* __restrict__ A2,
                                                        const float* __restrict__ W2,
                                                        const float* __restrict__ bias,
                                                        float* __restrict__ out) {
    const int lane = threadIdx.x & 31;
    const int wave = threadIdx.x >> 5;
    const int li   = lane & 15;          // row (A) / col (B,C) within tile
    const int kk   = (lane >> 4) << 1;   // lane half selects K {0,1} vs {2,3}
    const int m0   = blockIdx.x * GEMM_ROWS;
    const int n0   = wave * 32;

    const float* a1 = A1 + (size_t)(m0 + li) * D + kk;
    const float* a2 = A2 + (size_t)(m0 + li) * D + kk;
    const float* b1 = W1 + (size_t)(n0 + li) * D + kk;
    const float* b2 = W2 + (size_t)(n0 + li) * D + kk;

    v8f acc[5][2];
#pragma unroll
    for (int s = 0; s < 5; ++s) { acc[s][0] = v8f{}; acc[s][1] = v8f{}; }

    for (int k0 = 0; k0 < D; k0 += 4) {
        // B fragments for this wave's two column tiles (both weight matrices)
        v2f bL0 = *(const v2f*)(b1 + k0);
        v2f bL1 = *(const v2f*)(b1 + 16 * D + k0);
        v2f bR0 = *(const v2f*)(b2 + k0);
        v2f bR1 = *(const v2f*)(b2 + 16 * D + k0);
#pragma unroll
        for (int s = 0; s < 5; ++s) {
            const int aoff = s * 16 * D;
            v2f aA = *(const v2f*)(a1 + aoff + k0);   // agg path
            v2f aX = *(const v2f*)(a2 + aoff + k0);   // self path
            acc[s][0] = __builtin_amdgcn_wmma_f32_16x16x4_f32(
                false, aA, false, bL0, (short)0, acc[s][0], false, false);
            acc[s][0] = __builtin_amdgcn_wmma_f32_16x16x4_f32(
                false, aX, false, bR0, (short)0, acc[s][0], false, false);
            acc[s][1] = __builtin_amdgcn_wmma_f32_16x16x4_f32(
                false, aA, false, bL1, (short)0, acc[s][1], false, false);
            acc[s][1] = __builtin_amdgcn_wmma_f32_16x16x4_f32(
                false, aX, false, bR1, (short)0, acc[s][1], false, false);
        }
    }

    // C/D layout: acc element v -> row base + v + 8*(lane>=16), col n0+t*16+li
    const int mh = (lane >> 4) << 3;
#pragma unroll
    for (int t = 0; t < 2; ++t) {
        const int col = n0 + t * 16 + li;
        const float bn = bias[col];
#pragma unroll
        for (int s = 0; s < 5; ++s) {
            const int rbase = m0 + s * 16 + mh;
#pragma unroll
            for (int v = 0; v < 8; ++v)
                out[(size_t)(rbase + v) * D + col] = acc[s][t][v] + bn;
        }
    }
}

// ---------------- BN column stats (sum, sumsq) ------------------------------
__global__ __launch_bounds__(256) void col_stats_kernel(const float* __restrict__ h,
                                                        float* __restrict__ csum,
                                                        float* __restrict__ csq) {
    int t = threadIdx.x;                  // column
    float s = 0.0f, q = 0.0f;
    for (int r = blockIdx.x; r < N_NODES; r += gridDim.x) {
        float v = h[(size_t)r * D + t];
        s += v; q += v * v;
    }
    atomicAdd(&csum[t], s);
    atomicAdd(&csq[t], q);
}

__global__ __launch_bounds__(256) void bn_finalize_kernel(const float* __restrict__ csum,
                                                          const float* __restrict__ csq,
                                                          const float* __restrict__ gamma,
                                                          const float* __restrict__ beta,
                                                          float* __restrict__ bscale,
                                                          float* __restrict__ bshift) {
    int t = threadIdx.x;
    float mu  = csum[t] / (float)N_NODES;
    float var = csq[t] / (float)N_NODES - mu * mu;
    float rs  = rsqrtf(var + BN_EPS);
    float sc  = gamma[t] * rs;
    bscale[t] = sc;
    bshift[t] = beta[t] - mu * sc;
}

__global__ __launch_bounds__(256) void bn_relu_kernel(float* __restrict__ h,
                                                      const float* __restrict__ bscale,
                                                      const float* __restrict__ bshift) {
    int idx = blockIdx.x * 256 + threadIdx.x;   // float4 group id
    int c = (idx & 63) << 2;                    // column of first component
    float4 v = ((float4*)h)[idx];
    v.x = fmaxf(v.x * bscale[c + 0] + bshift[c + 0], 0.0f);
    v.y = fmaxf(v.y * bscale[c + 1] + bshift[c + 1], 0.0f);
    v.z = fmaxf(v.z * bscale[c + 2] + bshift[c + 2], 0.0f);
    v.w = fmaxf(v.w * bscale[c + 3] + bshift[c + 3], 0.0f);
    ((float4*)h)[idx] = v;
}

// ---------------- row-wise L2 normalize (in place) --------------------------
__global__ __launch_bounds__(256) void l2norm_kernel(float* __restrict__ h) {
    __shared__ float red[256];
    int r = blockIdx.x, t = threadIdx.x;
    float v = h[(size_t)r * D + t];
    red[t] = v * v;
    __syncthreads();
    for (int s = 128; s > 0; s >>= 1) {
        if (t < s) red[t] += red[t + s];
        __syncthreads();
    }
    float inv = 1.0f / fmaxf(sqrtf(red[0]), NRM_EPS);
    h[(size_t)r * D + t] = v * inv;
}

// ---------------- column sums of node embeddings ----------------------------
__global__ __launch_bounds__(256) void col_sum_kernel(const float* __restrict__ h,
                                                      float* __restrict__ gsum) {
    int t = threadIdx.x;
    float s = 0.0f;
    for (int r = blockIdx.x; r < N_NODES; r += gridDim.x)
        s += h[(size_t)r * D + t];
    atomicAdd(&gsum[t], s);
}

// ---------------- graph embedding: normalize(mean) --------------------------
__global__ __launch_bounds__(256) void pool_norm_kernel(const float* __restrict__ gsum,
                                                        float* __restrict__ gout) {
    __shared__ float red[256];
    int t = threadIdx.x;
    float g = gsum[t] / (float)N_NODES;
    red[t] = g * g;
    __syncthreads();
    for (int s = 128; s > 0; s >>= 1) {
        if (t < s) red[t] += red[t + s];
        __syncthreads();
    }
    float inv = 1.0f / fmaxf(sqrtf(red[0]), NRM_EPS);
    gout[t] = g * inv;
}

// ---------------------------------------------------------------------------
extern "C" void kernel_launch(void* const* d_in, const int* in_sizes, int n_in,
                              void* d_out, int out_size, void* d_ws, size_t ws_size,
                              hipStream_t stream) {
    const float* x     = (const float*)d_in[0];
    const float* W1l   = (const float*)d_in[1];
    const float* b1    = (const float*)d_in[2];
    const float* W1r   = (const float*)d_in[3];
    const float* W2l   = (const float*)d_in[4];
    const float* b2    = (const float*)d_in[5];
    const float* W2r   = (const float*)d_in[6];
    const float* gamma = (const float*)d_in[7];
    const float* beta  = (const float*)d_in[8];
    const int*   ei    = (const int*)d_in[9];
    (void)in_sizes; (void)n_in; (void)out_size; (void)ws_size;

    // workspace layout (floats)
    float* s      = (float*)d_ws;                 // [N,D] scatter accum / agg
    float* h1     = s      + (size_t)N_NODES * D; // [N,D] layer-1 activations
    float* cnt    = h1     + (size_t)N_NODES * D; // [N]
    float* csum   = cnt    + N_NODES;             // [D]
    float* csq    = csum   + D;                   // [D]
    float* bscale = csq    + D;                   // [D]
    float* bshift = bscale + D;                   // [D]
    float* gsum   = bshift + D;                   // [D]

    float* node_out  = (float*)d_out;                       // [N,D]
    float* graph_out = node_out + (size_t)N_NODES * D;      // [D]

    const int elem4_blocks = (N_NODES * D / 4) / 256;       // 12500
    const int gemm_blocks  = N_NODES / GEMM_ROWS;           // 625

    // deterministic zero-init of accumulators (graph-capture safe)
    hipMemsetAsync(s,   0, (size_t)N_NODES * D * sizeof(float), stream);
    hipMemsetAsync(cnt, 0, (size_t)(N_NODES + 5 * D) * sizeof(float), stream);

    // ---- layer 1 -----------------------------------------------------------
    count_kernel  <<<N_EDGES / 256, 256, 0, stream>>>(ei, cnt);
    scatter_kernel<<<N_EDGES / 4,   256, 0, stream>>>(x, ei, s);
    agg_kernel    <<<elem4_blocks,  256, 0, stream>>>(s, cnt);
    sage_gemm_kernel<<<gemm_blocks, 256, 0, stream>>>(s, W1l, x, W1r, b1, h1);
    col_stats_kernel<<<256, 256, 0, stream>>>(h1, csum, csq);
    bn_finalize_kernel<<<1, 256, 0, stream>>>(csum, csq, gamma, beta, bscale, bshift);
    bn_relu_kernel<<<elem4_blocks, 256, 0, stream>>>(h1, bscale, bshift);

    // ---- layer 2 -----------------------------------------------------------
    hipMemsetAsync(s, 0, (size_t)N_NODES * D * sizeof(float), stream);
    scatter_kernel<<<N_EDGES / 4,  256, 0, stream>>>(h1, ei, s);
    agg_kernel    <<<elem4_blocks, 256, 0, stream>>>(s, cnt);
    sage_gemm_kernel<<<gemm_blocks, 256, 0, stream>>>(s, W2l, h1, W2r, b2, node_out);

    // ---- normalize + pool --------------------------------------------------
    l2norm_kernel  <<<N_NODES, 256, 0, stream>>>(node_out);
    col_sum_kernel <<<256, 256, 0, stream>>>(node_out, gsum);
    pool_norm_kernel<<<1, 256, 0, stream>>>(gsum, graph_out);
}